// ClassicWaveletModule_9517647528367
// MI455X (gfx1250) — compile-verified
//
#include <hip/hip_runtime.h>

typedef float v2f __attribute__((ext_vector_type(2)));
typedef float v4f __attribute__((ext_vector_type(4)));
typedef float v8f __attribute__((ext_vector_type(8)));

#define LROW   16384
#define NROWS  2048
#define CHUNK  2048
#define NCHUNK (LROW / CHUNK)     // 8
#define XVALID 2168               // CHUNK + 56 halo left + 64 halo right
#define XLEN   2176               // padded
#define SIMN   512

__constant__ float c_dec_lo[8] = {
    -0.010597401785069032f,  0.032883011666982945f,  0.030841381835986965f,
    -0.18703481171888114f,  -0.02798376941698385f,   0.6308807679295904f,
     0.7148465705525415f,    0.23037781330885523f };
__constant__ float c_rec_lo[8] = {
     0.23037781330885523f,   0.7148465705525415f,    0.6308807679295904f,
    -0.02798376941698385f,  -0.18703481171888114f,   0.030841381835986965f,
     0.032883011666982945f, -0.010597401785069032f };

// ---------------------------------------------------------------------------
// Prep kernel: build the 16x128 composite low-path weight matrix W in d_ws by
// pushing 16 unit impulses through the exact interior dwt/idwt recurrences.
// low[16m + r] = sum_j W[r*128 + j] * x[16m + j - 56]
// ---------------------------------------------------------------------------
__global__ void wavelet_build_weights(float* __restrict__ W) {
    const int t = threadIdx.x;              // 16 threads, one impulse each
    for (int i = t; i < 16 * 128; i += 16) W[i] = 0.0f;
    __syncthreads();

    float A[SIMN], B[SIMN];
    const int q = 256 + t;                  // impulse position (deep interior)
    for (int i = 0; i < SIMN; ++i) A[i] = 0.0f;
    A[q] = 1.0f;

    auto dwt_lo = [](const float* src, float* dst) {
        for (int i = 0; i < SIMN; ++i) {
            float s = 0.0f;
            for (int tt = 0; tt < 8; ++tt) {
                int idx = 2 * i + tt - 6;
                if (idx >= 0 && idx < SIMN) s += src[idx] * c_dec_lo[7 - tt];
            }
            dst[i] = s;
        }
    };
    auto idwt_lo = [](const float* src, float* dst) {
        for (int p = 0; p < SIMN; ++p) dst[p] = 0.0f;
        for (int k = 0; k < SIMN; ++k) {
            float av = src[k];
            if (av != 0.0f) {
                for (int p = 2 * k - 6; p <= 2 * k + 1; ++p)
                    if (p >= 0 && p < SIMN) dst[p] += av * c_rec_lo[7 - (2 * k + 1 - p)];
            }
        }
    };

    dwt_lo(A, B);  dwt_lo(B, A);  dwt_lo(A, B);     // B = a3 impulse response
    idwt_lo(B, A); idwt_lo(A, B); idwt_lo(B, A);    // A = low impulse response

    for (int n = 0; n < SIMN; ++n) {
        float v = A[n];
        if (v != 0.0f) {
            int m = n >> 4, r = n & 15;
            int j = q - 16 * m + 56;                // OFS = 56
            if (j >= 0 && j < 128) W[r * 128 + j] = v;   // disjoint across threads
        }
    }
}

// ---------------------------------------------------------------------------
// Main kernel: per block, stage one 2048-sample row chunk (+halos) in LDS,
// each of 8 waves computes a 16x16 output tile (256 consecutive samples of
// `low`) via 32 steps of V_WMMA_F32_16X16X4_F32, then high = x - low.
// ---------------------------------------------------------------------------
__launch_bounds__(256)
__global__ void wavelet_lohi_wmma(const float* __restrict__ x,
                                  const float* __restrict__ Wg,
                                  float* __restrict__ lowp,
                                  float* __restrict__ highp) {
    __shared__ __align__(16) float wlds[16 * 128];
    __shared__ __align__(16) float xlds[XLEN];

    const int tid    = threadIdx.x;
    const int row    = (int)blockIdx.x >> 3;         // blockIdx = row*8 + chunk
    const int chunk  = (int)blockIdx.x & (NCHUNK - 1);
    const long roff  = (long)row * LROW;
    const int cstart = chunk * CHUNK;

    for (int i = tid; i < 16 * 128; i += 256) wlds[i] = Wg[i];
    for (int i = tid; i < XVALID; i += 256) {
        int g = cstart - 56 + i;
        if (g < 0)      g = -g - 1;                  // symmetric reflect
        if (g >= LROW)  g = 2 * LROW - 1 - g;
        xlds[i] = x[roff + g];
    }
    __syncthreads();

    const int wave  = tid >> 5;
    const int lane  = tid & 31;
    const int c     = lane & 15;                     // B column / D column (N)
    const int half  = lane >> 4;
    // A fragment: K = vgpr + 2*half, row M = lane&15 (16x4 f32 layout)
    const float* wrow = &wlds[c * 128 + 2 * half];
    // B fragment: B[k][c] = x[16*(m0+c) + k - 56]; LDS idx = wave*256 + 16c + k
    const float* xrow = &xlds[wave * 256 + 16 * c + 2 * half];

    v8f acc = {};
#pragma unroll
    for (int s = 0; s < 32; ++s) {                   // K = 128 in steps of 4
        v2f a = *reinterpret_cast<const v2f*>(wrow + 4 * s);
        v2f b = *reinterpret_cast<const v2f*>(xrow + 4 * s);
        acc = __builtin_amdgcn_wmma_f32_16x16x4_f32(
            /*neg_a=*/false, a, /*neg_b=*/false, b,
            /*c_mod=*/(short)0, acc, /*reuse_a=*/false, /*reuse_b=*/false);
    }

    // Lane (c, half) holds D rows M = i + 8*half, col c -> 8 consecutive outputs
    const int n0   = cstart + wave * 256 + 16 * c + 8 * half;
    const int lidx = wave * 256 + 16 * c + 8 * half + 56;   // x in LDS for same n

    v4f xa = *reinterpret_cast<const v4f*>(&xlds[lidx]);
    v4f xb = *reinterpret_cast<const v4f*>(&xlds[lidx + 4]);

    v4f lo0 = { acc[0], acc[1], acc[2], acc[3] };
    v4f lo1 = { acc[4], acc[5], acc[6], acc[7] };
    v4f hi0 = { xa.x - lo0.x, xa.y - lo0.y, xa.z - lo0.z, xa.w - lo0.w };
    v4f hi1 = { xb.x - lo1.x, xb.y - lo1.y, xb.z - lo1.z, xb.w - lo1.w };

    *reinterpret_cast<v4f*>(lowp  + roff + n0)     = lo0;
    *reinterpret_cast<v4f*>(lowp  + roff + n0 + 4) = lo1;
    *reinterpret_cast<v4f*>(highp + roff + n0)     = hi0;
    *reinterpret_cast<v4f*>(highp + roff + n0 + 4) = hi1;
}

extern "C" void kernel_launch(void* const* d_in, const int* in_sizes, int n_in,
                              void* d_out, int out_size, void* d_ws, size_t ws_size,
                              hipStream_t stream) {
    (void)in_sizes; (void)n_in; (void)out_size; (void)ws_size;
    const float* x  = (const float*)d_in[0];
    float* W        = (float*)d_ws;                       // 16*128 floats (8 KB)
    float* lowp     = (float*)d_out;
    float* highp    = lowp + (size_t)NROWS * LROW;

    hipLaunchKernelGGL(wavelet_build_weights, dim3(1), dim3(16), 0, stream, W);
    hipLaunchKernelGGL(wavelet_lohi_wmma, dim3(NROWS * NCHUNK), dim3(256), 0, stream,
                       x, W, lowp, highp);
}